// STGCN_64192581206791
// MI455X (gfx1250) — compile-verified
//
#include <hip/hip_runtime.h>

typedef float v2f __attribute__((ext_vector_type(2)));
typedef float v8f __attribute__((ext_vector_type(8)));
typedef int v4i __attribute__((vector_size(16))); // matches builtin param type

#define ASG __attribute__((address_space(1)))
#define ASL __attribute__((address_space(3)))

#if defined(__gfx1250__) && __has_builtin(__builtin_amdgcn_global_load_async_to_lds_b128) && \
    __has_builtin(__builtin_amdgcn_s_wait_asynccnt)
#define USE_ASYNC_LDS 1
#else
#define USE_ASYNC_LDS 0
#endif

static __device__ __forceinline__ v8f wmma_f32(v2f a, v2f b, v8f c) {
  // V_WMMA_F32_16X16X4_F32: D = A(16x4) * B(4x16) + C(16x16), full fp32
  return __builtin_amdgcn_wmma_f32_16x16x4_f32(false, a, false, b, (short)0, c,
                                               false, false);
}

namespace {
constexpr int Nn = 64, Cc = 64, Tt = 300, Vv = 25, Kk = 3, Ff = 64, FK = 192;
constexpr float EPSv = 1e-3f;
constexpr int TB = 2;        // t's per workgroup in stage 2
constexpr int HALO = TB + 8; // temporal halo rows (kernel size 9, pad 4)
} // namespace

// ---------------------------------------------------------------------------
// Stage 1: fused 1x1 conv (channel GEMM) + graph contraction + BN1 + ReLU
// One workgroup per (n, t); 128 threads = 4 waves.
//   GEMM1: P[fk][v] = sum_c W1[c][fk] * x[n][c][t][v]  (M=192, N=32pad, K=64)
//   GEMM2: Z[f][w]  = sum_{k,v} (P[k*F+f][v]+b1) * A[k][t][v][w] (M=64,N=32,K=76pad)
//   store z[n][t][w][f] = relu(BN1(Z))
// ---------------------------------------------------------------------------
__global__ __launch_bounds__(128)
void stgcn_fused_gcn(const float* __restrict__ x, const float* __restrict__ W1,
                     const float* __restrict__ bias1, const float* __restrict__ A,
                     const float* __restrict__ gamma1, const float* __restrict__ beta1,
                     const float* __restrict__ mean1, const float* __restrict__ var1,
                     float* __restrict__ z) {
  __shared__ float Xs[Cc * 32];  // x tile  [c][v], v padded 25->32
  __shared__ float Ps[FK * 32];  // GEMM1 out (+conv bias) [fk][v]
  __shared__ float As[80 * 32];  // adjacency [k*25+v][w], padded 75->80, 25->32

  const int nt = blockIdx.x;
  const int n = nt / Tt, t = nt - n * Tt;
  const int tid = threadIdx.x;
  const int lane = tid & 31;
  const int wv = tid >> 5;
  const int l16 = lane & 15; // column / M-row lane id
  const int lh = lane >> 4;  // lane half (K striping)

  for (int i = tid; i < Cc * 32; i += 128) {
    const int c = i >> 5, v = i & 31;
    Xs[i] = (v < Vv) ? x[((n * Cc + c) * Tt + t) * Vv + v] : 0.f;
  }
  for (int i = tid; i < 80 * 32; i += 128) {
    const int kv = i >> 5, w = i & 31;
    float val = 0.f;
    if (kv < Kk * Vv && w < Vv) {
      const int k = kv / Vv, v = kv - k * Vv;
      val = A[((k * Tt + t) * Vv + v) * Vv + w];
    }
    As[i] = val;
  }
  __syncthreads();

  // ---- GEMM1: A-operand = W1^T streamed from L2, B-operand = Xs (LDS) ----
  for (int mt = wv; mt < 12; mt += 4) { // 12 M-tiles of 16 over fk=0..191
    v8f acc0 = {}, acc1 = {};
    const int m = mt * 16 + l16;
#pragma unroll 4
    for (int ks = 0; ks < 16; ++ks) { // K = 64 in steps of 4
      const int k0 = ks * 4 + lh * 2;
      v2f a, b0, b1;
      a.x = W1[k0 * FK + m];
      a.y = W1[(k0 + 1) * FK + m];
      b0.x = Xs[k0 * 32 + l16];
      b0.y = Xs[(k0 + 1) * 32 + l16];
      b1.x = Xs[k0 * 32 + 16 + l16];
      b1.y = Xs[(k0 + 1) * 32 + 16 + l16];
      acc0 = wmma_f32(a, b0, acc0);
      acc1 = wmma_f32(a, b1, acc1);
    }
#pragma unroll
    for (int r = 0; r < 8; ++r) { // D: M = mt*16 + lh*8 + r, N = l16 / 16+l16
      const int fk = mt * 16 + lh * 8 + r;
      const float bb = bias1[fk]; // fold conv bias here
      Ps[fk * 32 + l16] = acc0[r] + bb;
      Ps[fk * 32 + 16 + l16] = acc1[r] + bb;
    }
  }
  __syncthreads();

  // ---- GEMM2: A-operand = Ps (LDS), B-operand = As (LDS) ----
  {
    const int mt = wv; // 4 waves <-> 4 M-tiles over f=0..63
    const int m = mt * 16 + l16;
    v8f acc0 = {}, acc1 = {};
    for (int ks = 0; ks < 19; ++ks) { // K = 75 padded to 76
      const int k0 = ks * 4 + lh * 2;
      const int k1 = k0 + 1;
      v2f a, b0, b1;
      a.x = (k0 < 75) ? Ps[((k0 / Vv) * Ff + m) * 32 + (k0 % Vv)] : 0.f;
      a.y = (k1 < 75) ? Ps[((k1 / Vv) * Ff + m) * 32 + (k1 % Vv)] : 0.f;
      b0.x = As[k0 * 32 + l16];
      b0.y = As[k1 * 32 + l16];
      b1.x = As[k0 * 32 + 16 + l16];
      b1.y = As[k1 * 32 + 16 + l16];
      acc0 = wmma_f32(a, b0, acc0);
      acc1 = wmma_f32(a, b1, acc1);
    }
    // epilogue: BN1 + ReLU, store z[n][t][w][f] (f-contiguous)
#pragma unroll
    for (int r = 0; r < 8; ++r) {
      const int f = mt * 16 + lh * 8 + r;
      const float rstd = rsqrtf(var1[f] + EPSv);
      const float s = gamma1[f] * rstd;
      const float bb = beta1[f] - mean1[f] * s;
      const float z0 = fmaxf(acc0[r] * s + bb, 0.f);
      const float z1 = fmaxf(acc1[r] * s + bb, 0.f);
      z[(nt * Vv + l16) * Ff + f] = z0; // w = l16 < 25 always
      if (16 + l16 < Vv) z[(nt * Vv + 16 + l16) * Ff + f] = z1;
    }
  }
}

// ---------------------------------------------------------------------------
// Stage 2: fused temporal conv [9,1] SAME (implicit GEMM, K = 9*64 = 576)
//          + BN2 + residual(+x) + ReLU.
// One workgroup per (n, 2 consecutive t); 128 threads = 4 waves.
// Each wave owns one fixed f-tile -> one Wt B-fragment feeds 4 WMMAs.
// z-halo staged to LDS via async global->LDS (ASYNCcnt) when available.
// ---------------------------------------------------------------------------
__global__ __launch_bounds__(128)
void stgcn_fused_tcn(const float* __restrict__ z, const float* __restrict__ Wt,
                     const float* __restrict__ bt,
                     const float* __restrict__ gamma2, const float* __restrict__ beta2,
                     const float* __restrict__ mean2, const float* __restrict__ var2,
                     const float* __restrict__ x, float* __restrict__ out) {
  __shared__ alignas(16) float zs[HALO * Vv * Ff]; // [tloc][v][g], 64000 B

  const int ntb = (Tt + TB - 1) / TB; // 150
  const int n = blockIdx.x / ntb;
  const int t0 = (blockIdx.x - n * ntb) * TB;
  const int tid = threadIdx.x;
  const int lane = tid & 31;
  const int wv = tid >> 5;
  const int l16 = lane & 15;
  const int lh = lane >> 4;

#if USE_ASYNC_LDS
  // stage the 10-row t-halo with async B128 copies (no VGPR round-trip)
  for (int u = tid; u < HALO * Vv * (Ff / 4); u += 128) { // 16-byte units
    const int tl = u / (Vv * Ff / 4);
    const int rem = u - tl * (Vv * Ff / 4);
    const int tp = t0 - 4 + tl;
    float* lp = &zs[u * 4];
    if (tp >= 0 && tp < Tt) {
      const float* gp = &z[(n * Tt + tp) * (Vv * Ff) + rem * 4];
      __builtin_amdgcn_global_load_async_to_lds_b128((ASG v4i*)gp,
                                                     (ASL v4i*)lp, 0, 0);
    } else {
      *(float4*)lp = make_float4(0.f, 0.f, 0.f, 0.f); // zero the SAME halo
    }
  }
  __builtin_amdgcn_s_wait_asynccnt(0);
#else
  for (int i = tid; i < HALO * Vv * Ff; i += 128) {
    const int tl = i / (Vv * Ff);
    const int rem = i - tl * (Vv * Ff);
    const int tp = t0 - 4 + tl;
    zs[i] = (tp >= 0 && tp < Tt) ? z[(n * Tt + tp) * (Vv * Ff) + rem] : 0.f;
  }
#endif
  __syncthreads();

  // wave -> fixed f-tile; 4 accumulators (2 t x 2 v-tiles) share each B-frag
  const int ft = wv;
  const int f = ft * 16 + l16;
  const int v0 = l16;      // v-tile 0 row (always < 25)
  const int v1 = 16 + l16; // v-tile 1 row (masked beyond 25)
  v8f acc[TB][2];
#pragma unroll
  for (int tl = 0; tl < TB; ++tl) {
    acc[tl][0] = (v8f){};
    acc[tl][1] = (v8f){};
  }

  for (int ks = 0; ks < 144; ++ks) { // K = 576 in steps of 4
    const int k0 = ks * 4 + lh * 2;  // kidx = kt*64 + g
    const int k1 = k0 + 1;
    v2f b;
    b.x = Wt[k0 * Ff + f]; // Wt[kt][0][g][f] flat = kidx*64 + f (L2 resident)
    b.y = Wt[k1 * Ff + f];
    const int r0 = k0 >> 6, g0 = k0 & 63;
    const int r1 = k1 >> 6, g1 = k1 & 63;
#pragma unroll
    for (int tl = 0; tl < TB; ++tl) {
      v2f a0, a1;
      a0.x = zs[((tl + r0) * Vv + v0) * Ff + g0];
      a0.y = zs[((tl + r1) * Vv + v0) * Ff + g1];
      a1.x = (v1 < Vv) ? zs[((tl + r0) * Vv + v1) * Ff + g0] : 0.f;
      a1.y = (v1 < Vv) ? zs[((tl + r1) * Vv + v1) * Ff + g1] : 0.f;
      acc[tl][0] = wmma_f32(a0, b, acc[tl][0]);
      acc[tl][1] = wmma_f32(a1, b, acc[tl][1]);
    }
  }

  // epilogue: BN2 (conv bias folded) + residual + ReLU
  const float rstd = rsqrtf(var2[f] + EPSv);
  const float s = gamma2[f] * rstd;
  const float bb = beta2[f] - mean2[f] * s + bt[f] * s;
#pragma unroll
  for (int tl = 0; tl < TB; ++tl) {
    const int t = t0 + tl;
    if (t < Tt) {
#pragma unroll
      for (int mt = 0; mt < 2; ++mt) {
#pragma unroll
        for (int r = 0; r < 8; ++r) {
          const int vr = mt * 16 + lh * 8 + r;
          if (vr < Vv) {
            const int oi = ((n * Cc + f) * Tt + t) * Vv + vr;
            out[oi] = fmaxf(acc[tl][mt][r] * s + bb + x[oi], 0.f);
          }
        }
      }
    }
  }
}

// ---------------------------------------------------------------------------
extern "C" void kernel_launch(void* const* d_in, const int* in_sizes, int n_in,
                              void* d_out, int out_size, void* d_ws, size_t ws_size,
                              hipStream_t stream) {
  const float* x      = (const float*)d_in[0];
  const float* W1     = (const float*)d_in[1];
  const float* b1     = (const float*)d_in[2];
  const float* A      = (const float*)d_in[3];
  const float* gamma1 = (const float*)d_in[4];
  const float* beta1  = (const float*)d_in[5];
  const float* mean1  = (const float*)d_in[6];
  const float* var1   = (const float*)d_in[7];
  const float* Wt     = (const float*)d_in[8];
  const float* bt     = (const float*)d_in[9];
  const float* gamma2 = (const float*)d_in[10];
  const float* beta2  = (const float*)d_in[11];
  const float* mean2  = (const float*)d_in[12];
  const float* var2   = (const float*)d_in[13];

  float* z = (float*)d_ws; // N*T*V*F fp32 intermediate (~117 MB)

  stgcn_fused_gcn<<<Nn * Tt, 128, 0, stream>>>(x, W1, b1, A, gamma1, beta1,
                                               mean1, var1, z);
  stgcn_fused_tcn<<<Nn * ((Tt + TB - 1) / TB), 128, 0, stream>>>(
      z, Wt, bt, gamma2, beta2, mean2, var2, x, (float*)d_out);
}